// PillarQueryAndGroup_25958782337389
// MI455X (gfx1250) — compile-verified
//
#include <hip/hip_runtime.h>
#include <hip/hip_bf16.h>
#include <stdint.h>

// ---------------------------------------------------------------------------
// PillarQueryAndGroup for MI455X (gfx1250).
// Memory-bound: ~310 MB of traffic -> ~13-15 us at 23.3 TB/s HBM. No matrix
// math exists in this op, so no WMMA; we instead exercise the CDNA5 async
// global->LDS path (ASYNCcnt) for feature staging and global_prefetch_b8.
// Occupancy mask is bit-packed (1 MB) and compaction ranks come from a
// two-level prefix sum over per-word popcounts.
// ---------------------------------------------------------------------------

#define GRID_H 1440
#define GRID_W 1440
#define PSZ    0.075f
#define XYOFF  (-53.9625f)   // PILLAR_SIZE/2 + PC_RANGE[0] (== y offset too)

#if defined(__HIP_DEVICE_COMPILE__) && \
    __has_builtin(__builtin_amdgcn_global_load_async_to_lds_b128) && \
    __has_builtin(__builtin_amdgcn_s_wait_asynccnt)
#define USE_ASYNC_LDS 1
#else
#define USE_ASYNC_LDS 0
#endif

typedef int v4i __attribute__((vector_size(4 * sizeof(int))));
typedef __attribute__((address_space(1))) v4i* gv4i_ptr;   // global (AS1) int4*
typedef __attribute__((address_space(3))) v4i* lv4i_ptr;   // LDS (AS3) int4*

// ---------------- helpers ----------------

__device__ __forceinline__ unsigned block_excl_scan_256(unsigned v, unsigned* lds, int t) {
    // Hillis-Steele inclusive scan over 256 lanes worth of per-thread sums.
    unsigned run = v;
    lds[t] = run;
    __syncthreads();
    #pragma unroll
    for (int off = 1; off < 256; off <<= 1) {
        unsigned y = (t >= off) ? lds[t - off] : 0u;
        __syncthreads();
        run += y;
        lds[t] = run;
        __syncthreads();
    }
    return run - v;   // exclusive
}

__device__ __forceinline__ int batch_of(int i, const int* __restrict__ cnts, int batch) {
    int b = 0, acc = 0;
    for (int k = 0; k < batch - 1; ++k) { acc += cnts[k]; b += (i >= acc) ? 1 : 0; }
    return b;
}

// ---------------- kernels ----------------

__global__ void __launch_bounds__(256) pqg_zero(unsigned* __restrict__ p, int n) {
    int i0 = (blockIdx.x * 256 + threadIdx.x) * 4;
    if (i0 + 3 < n) {
        *(uint4*)(p + i0) = make_uint4(0u, 0u, 0u, 0u);
    } else {
        for (int k = 0; k < 4; ++k) if (i0 + k < n) p[i0 + k] = 0u;
    }
}

__global__ void __launch_bounds__(256) pqg_scatter(const int* __restrict__ pts_xy,
                                                   const int* __restrict__ cnts,
                                                   unsigned* __restrict__ mask,
                                                   int N, int batch) {
    int i = blockIdx.x * 256 + threadIdx.x;
    if (i >= N) return;
    // stream-ahead prefetch of upcoming xy pairs (global_prefetch_b8)
    __builtin_prefetch(pts_xy + 2 * (size_t)min(i + 16384, N - 1), 0, 1);
    int x = pts_xy[2 * i + 0];
    int y = pts_xy[2 * i + 1];
    int b = batch_of(i, cnts, batch);
    unsigned flat = ((unsigned)(b * GRID_H + y)) * (unsigned)GRID_W + (unsigned)x;
    atomicOr(&mask[flat >> 5], 1u << (flat & 31u));
}

__global__ void __launch_bounds__(256) pqg_reduce(const unsigned* __restrict__ mask,
                                                  unsigned* __restrict__ blockSums,
                                                  int nWords) {
    __shared__ unsigned lds[256];
    int t  = threadIdx.x;
    int w0 = blockIdx.x * 1024 + t * 4;
    unsigned s = 0;
    if (w0 + 3 < nWords) {
        uint4 v = *(const uint4*)(mask + w0);
        s = __popc(v.x) + __popc(v.y) + __popc(v.z) + __popc(v.w);
    } else {
        for (int k = 0; k < 4; ++k) if (w0 + k < nWords) s += __popc(mask[w0 + k]);
    }
    lds[t] = s;
    __syncthreads();
    for (int off = 128; off > 0; off >>= 1) {
        if (t < off) lds[t] += lds[t + off];
        __syncthreads();
    }
    if (t == 0) blockSums[blockIdx.x] = lds[0];
}

__global__ void __launch_bounds__(256) pqg_scan_blocks(unsigned* __restrict__ blockSums, int nB) {
    __shared__ unsigned lds[256];
    int t = threadIdx.x;
    int k = (nB + 255) / 256;            // contiguous elems per thread (==1 here)
    unsigned loc[16];
    unsigned s = 0;
    #pragma unroll
    for (int j = 0; j < 16; ++j) {
        if (j >= k) break;
        int idx = t * k + j;
        unsigned v = (idx < nB) ? blockSums[idx] : 0u;
        loc[j] = v;
        s += v;
    }
    unsigned excl = block_excl_scan_256(s, lds, t);
    #pragma unroll
    for (int j = 0; j < 16; ++j) {
        if (j >= k) break;
        int idx = t * k + j;
        if (idx < nB) { unsigned v = loc[j]; blockSums[idx] = excl; excl += v; }
    }
}

__global__ void __launch_bounds__(256) pqg_scan_emit(const unsigned* __restrict__ mask,
                                                     const unsigned* __restrict__ blockOffs,
                                                     unsigned* __restrict__ wprefix,
                                                     float* __restrict__ out,
                                                     int nWords) {
    __shared__ unsigned lds[256];
    int t  = threadIdx.x;
    int w0 = blockIdx.x * 1024 + t * 4;
    unsigned mv[4];
    unsigned s = 0;
    #pragma unroll
    for (int k = 0; k < 4; ++k) {
        int w = w0 + k;
        mv[k] = (w < nWords) ? mask[w] : 0u;
        s += __popc(mv[k]);
    }
    unsigned excl = block_excl_scan_256(s, lds, t) + blockOffs[blockIdx.x];
    #pragma unroll
    for (int k = 0; k < 4; ++k) {
        int w = w0 + k;
        if (w >= nWords) break;
        wprefix[w] = excl;
        unsigned m    = mv[k];
        unsigned base = (unsigned)w * 32u;
        while (m) {
            unsigned bpos = (unsigned)__ffs((int)m) - 1u;
            unsigned flat = base + bpos;
            unsigned xx  = flat % (unsigned)GRID_W;
            unsigned tmp = flat / (unsigned)GRID_W;
            unsigned yy  = tmp % (unsigned)GRID_H;
            unsigned bb  = tmp / (unsigned)GRID_H;
            float* o = out + (size_t)excl * 3u;   // pillar_indices[rank] = (b, y, x)
            o[0] = (float)bb;
            o[1] = (float)yy;
            o[2] = (float)xx;
            ++excl;
            m &= (m - 1u);
        }
    }
}

__global__ void __launch_bounds__(256) pqg_points(const int* __restrict__ pts_xy,
                                                  const int* __restrict__ cnts,
                                                  const float* __restrict__ feats,
                                                  const unsigned* __restrict__ mask,
                                                  const unsigned* __restrict__ wprefix,
                                                  float* __restrict__ out,
                                                  int N, int batch, long long P3) {
    __shared__ __align__(16) float tile[256 * 16];   // 16 KB feature tile
    const int t          = threadIdx.x;
    const int blockStart = blockIdx.x * 256;
    const int validPts   = min(256, N - blockStart);
    const int nVec       = validPts * 4;             // float4 count in tile
    const float4* g = (const float4*)feats + (size_t)blockStart * 4;
    float4*       l = (float4*)tile;

#if USE_ASYNC_LDS
    // Coalesced async global->LDS copy: each lane issues b128 async loads;
    // tracked on ASYNCcnt, waited before the block barrier.
    #pragma unroll
    for (int j = 0; j < 4; ++j) {
        int v = j * 256 + t;
        if (v < nVec) {
            __builtin_amdgcn_global_load_async_to_lds_b128(
                (gv4i_ptr)(g + v),
                (lv4i_ptr)(l + v),
                0, 0);
        }
    }
    __builtin_amdgcn_s_wait_asynccnt(0);
#else
    #pragma unroll
    for (int j = 0; j < 4; ++j) {
        int v = j * 256 + t;
        if (v < nVec) l[v] = g[v];
    }
#endif
    __syncthreads();

    int i = blockStart + t;
    if (i < N) {
        __builtin_prefetch(pts_xy + 2 * (size_t)min(i + 8192, N - 1), 0, 1);
        int x = pts_xy[2 * i + 0];
        int y = pts_xy[2 * i + 1];
        int b = batch_of(i, cnts, batch);
        unsigned flat = ((unsigned)(b * GRID_H + y)) * (unsigned)GRID_W + (unsigned)x;
        unsigned w    = flat >> 5;
        unsigned bit  = flat & 31u;
        unsigned rank = wprefix[w] + (unsigned)__popc(mask[w] & ((1u << bit) - 1u));

        out[P3 + i] = (float)rank;                         // point_pillar_indices

        // pillar center of the point's own cell; match JAX's separate mul+add rounding
        float cx = __fadd_rn(__fmul_rn((float)x, PSZ), XYOFF);
        float cy = __fadd_rn(__fmul_rn((float)y, PSZ), XYOFF);

        const float* f  = tile + t * 16;
        float* row = out + P3 + (long long)N + 18LL * i;   // point_pillar_features row
        row[0] = __fsub_rn(f[0], cx);
        row[1] = __fsub_rn(f[1], cy);
        #pragma unroll
        for (int c = 0; c < 16; ++c) row[2 + c] = f[c];
    }
}

// ---------------- host launcher ----------------

extern "C" void kernel_launch(void* const* d_in, const int* in_sizes, int n_in,
                              void* d_out, int out_size, void* d_ws, size_t ws_size,
                              hipStream_t stream) {
    const int*   pts_xy = (const int*)d_in[0];
    const int*   cnts   = (const int*)d_in[1];
    const float* feats  = (const float*)d_in[2];
    float*       out    = (float*)d_out;

    const int N     = in_sizes[0] / 2;
    const int batch = in_sizes[1];

    const long long BHW    = (long long)batch * GRID_H * GRID_W;
    const int nWords       = (int)((BHW + 31) / 32);       // 259,200 for 4x1440x1440
    const int nWordsAl     = (nWords + 3) & ~3;            // keep uint4 alignment
    const int nScanBlocks  = (nWords + 1023) / 1024;       // 254

    // workspace layout: [mask | wprefix | blockSums]  (~2.1 MB)
    unsigned* mask      = (unsigned*)d_ws;
    unsigned* wprefix   = mask + nWordsAl;
    unsigned* blockSums = wprefix + nWordsAl;

    // out_size = 3*P + N + 18*N  ->  P*3 = out_size - 19*N
    const long long P3 = (long long)out_size - 19LL * (long long)N;

    pqg_zero      <<<(nWords + 1023) / 1024, 256, 0, stream>>>(mask, nWords);
    pqg_scatter   <<<(N + 255) / 256,        256, 0, stream>>>(pts_xy, cnts, mask, N, batch);
    pqg_reduce    <<<nScanBlocks,            256, 0, stream>>>(mask, blockSums, nWords);
    pqg_scan_blocks<<<1,                     256, 0, stream>>>(blockSums, nScanBlocks);
    pqg_scan_emit <<<nScanBlocks,            256, 0, stream>>>(mask, blockSums, wprefix, out, nWords);
    pqg_points    <<<(N + 255) / 256,        256, 0, stream>>>(pts_xy, cnts, feats, mask, wprefix,
                                                               out, N, batch, P3);
}